// LightweightMambaEncoder_89129161327126
// MI455X (gfx1250) — compile-verified
//
#include <hip/hip_runtime.h>
#include <hip/hip_bf16.h>
#include <math.h>

// ---------------------------------------------------------------------------
// LightweightMambaEncoder on MI455X (gfx1250, wave32, WMMA)
//
// Per layer (fp32 throughout, matching reference numerics):
//   1) xz   = h @ in_proj_w^T              (WMMA f32 16x16x4 GEMM, N=512)
//   2) u    = silu(causal_dwconv(xz[:, :256]))
//   3) xdbl = u @ x_proj_w^T               (WMMA GEMM, N=40 padded to 48)
//   4) fused scan: delta=softplus(dt@dt_w^T+dt_b); s=exp(delta*A)*s+delta*u*B;
//      y=(sum s*C + u*D) * silu(z)         (1024 parallel channels, state in regs)
//   5) h    = y @ out_proj_w^T             (WMMA GEMM, N=128)
// Then mean-pool over L and latent projection.
// ---------------------------------------------------------------------------

#define B_SZ     4
#define L_SZ     2048
#define IN_DIM   64
#define DMODEL   128
#define DINNER   256
#define DSTATE   16
#define DTRANK   8
#define M_ROWS   (B_SZ * L_SZ)          // 8192
#define XDBL_LD  48                     // 40 padded to 48

typedef __attribute__((ext_vector_type(2))) float v2f;
typedef __attribute__((ext_vector_type(8))) float v8f;

// ---------------------------------------------------------------------------
// fp32 WMMA GEMM, fully compile-time shaped:  C[M_ROWS x NPAD] = A * B (+bias)
//   BT==1 : B[k][n] = Bw[n*K + k]   (weight stored [N, K])
//   BT==0 : B[k][n] = Bw[k*NREAL + n]
//   NREAL < NPAD -> zero-mask padded columns (x_proj 40->48)
//   NB    : adjacent 16x16 output tiles per wave (A fragment reused)
// K loop fully unrolled: all loads are global_load with immediate offsets off
// fixed base pointers; no address arithmetic, no branches, no FLAT path.
// A 16x4 layout: lane m = lane&15, K pair = (lane>>4)*2.
// C/D 16x16 layout: VGPR v -> M = v + (lane>>4)*8, N = lane&15.
// ---------------------------------------------------------------------------
template <int BT, int NREAL, int NPAD, int K, int NB, bool BIAS>
__global__ void gemm_f32_wmma(const float* __restrict__ A,
                              const float* __restrict__ Bw,
                              const float* __restrict__ bias,
                              float* __restrict__ C) {
  constexpr int GUARD = (NREAL != NPAD) ? 1 : 0;
  const int wave = (blockIdx.x * blockDim.x + threadIdx.x) >> 5;
  const int lane = threadIdx.x & 31;
  constexpr int NtB = (NPAD / 16) / NB;
  if (wave >= (M_ROWS / 16) * NtB) return;
  const int tm = wave / NtB, tn = wave % NtB;
  const int m0 = tm << 4, n0 = (tn * NB) << 4;

  const int mrow = lane & 15;          // A row held by this lane
  const int kk   = (lane >> 4) << 1;   // K sub-pair: 0 or 2
  const int ncol = lane & 15;          // B/C column held by this lane

  const float* ap = A + (size_t)(m0 + mrow) * K + kk;

  float bmask[NB];
  int   nidx[NB];      // column (clamped for padded lanes)
#pragma unroll
  for (int t = 0; t < NB; ++t) {
    const int n = n0 + 16 * t + ncol;
    const bool ok = (!GUARD) || (n < NREAL);
    bmask[t] = ok ? 1.0f : 0.0f;
    nidx[t]  = ok ? n : 0;
  }

  v8f acc[NB];
#pragma unroll
  for (int t = 0; t < NB; ++t) acc[t] = v8f{};

#pragma unroll
  for (int k0 = 0; k0 < K; k0 += 4) {
    v2f a;
    a.x = ap[k0];
    a.y = ap[k0 + 1];
#pragma unroll
    for (int t = 0; t < NB; ++t) {
      v2f b;
      if (BT) {
        const float* bp = Bw + (size_t)nidx[t] * K + kk;   // fixed base, imm offs
        b.x = bp[k0];
        b.y = bp[k0 + 1];
      } else {
        const float* bp = Bw + (size_t)kk * NREAL + nidx[t];
        b.x = bp[k0 * NREAL];
        b.y = bp[(k0 + 1) * NREAL];
      }
      if (GUARD) { b.x *= bmask[t]; b.y *= bmask[t]; }
      acc[t] = __builtin_amdgcn_wmma_f32_16x16x4_f32(
          /*neg_a=*/false, a, /*neg_b=*/false, b,
          /*c_mod=*/(short)0, acc[t], /*reuse_a=*/false, /*reuse_b=*/false);
    }
  }

  const int row_off = (lane >> 4) << 3;
#pragma unroll
  for (int t = 0; t < NB; ++t) {
    const int n = n0 + 16 * t + ncol;           // store true column (padded ok)
    const float bv = BIAS ? bias[nidx[t]] * bmask[t] : 0.0f;
    float* cp = C + (size_t)(m0 + row_off) * NPAD + n;
#pragma unroll
    for (int v = 0; v < 8; ++v)
      cp[v * NPAD] = acc[t][v] + bv;
  }
}

// ---------------------------------------------------------------------------
// Causal depthwise conv (width 4) + SiLU.  Input = first DINNER cols of xz.
// Lanes within a wave share (b,l) -> boundary branches are wave-uniform.
// ---------------------------------------------------------------------------
__global__ void conv_silu_kernel(const float* __restrict__ xz,   // [M, 2*DINNER]
                                 const float* __restrict__ cw,   // [DINNER, 4]
                                 const float* __restrict__ cb,   // [DINNER]
                                 float* __restrict__ u) {        // [M, DINNER]
  const int tid = blockIdx.x * blockDim.x + threadIdx.x;
  if (tid >= M_ROWS * DINNER) return;
  const int d  = tid & (DINNER - 1);
  const int bl = tid >> 8;                 // b*L + l
  const int l  = bl & (L_SZ - 1);

  float acc = cb[d];
#pragma unroll
  for (int j = 0; j < 4; ++j) {
    const int li = l - 3 + j;
    if (li >= 0)
      acc += cw[d * 4 + j] * xz[(size_t)(bl - 3 + j) * (2 * DINNER) + d];
  }
  u[tid] = acc / (1.0f + __expf(-acc));    // silu
}

// ---------------------------------------------------------------------------
// Fused selective scan. One thread per (b, d) channel; 16-wide state in regs.
// Fuses dt_proj + softplus, discretization, recurrence, C-readout, D-skip,
// and the silu(z) gate. Writes y [M, DINNER].
// ---------------------------------------------------------------------------
__global__ void scan_kernel(const float* __restrict__ xdbl,   // [M, XDBL_LD]
                            const float* __restrict__ u_buf,  // [M, DINNER]
                            const float* __restrict__ xz,     // [M, 2*DINNER]
                            const float* __restrict__ dt_w,   // [DINNER, DTRANK]
                            const float* __restrict__ dt_b,   // [DINNER]
                            const float* __restrict__ A_log,  // [DINNER, DSTATE]
                            const float* __restrict__ Dskip,  // [DINNER]
                            float* __restrict__ y_buf) {      // [M, DINNER]
  const int tid = blockIdx.x * blockDim.x + threadIdx.x;
  if (tid >= B_SZ * DINNER) return;
  const int d = tid & (DINNER - 1);
  const int b = tid >> 8;

  float An[DSTATE], s[DSTATE];
#pragma unroll
  for (int nn = 0; nn < DSTATE; ++nn) {
    An[nn] = -__expf(A_log[d * DSTATE + nn]);   // A = -exp(A_log)
    s[nn]  = 0.0f;
  }
  float dw[DTRANK];
#pragma unroll
  for (int r = 0; r < DTRANK; ++r) dw[r] = dt_w[d * DTRANK + r];
  const float db = dt_b[d];
  const float Dk = Dskip[d];

  const float* xr = xdbl + (size_t)b * L_SZ * XDBL_LD;
  const float* up = u_buf + (size_t)b * L_SZ * DINNER + d;
  const float* zp = xz + (size_t)b * L_SZ * (2 * DINNER) + DINNER + d;
  float*       yp = y_buf + (size_t)b * L_SZ * DINNER + d;

  for (int l = 0; l < L_SZ; ++l) {
    float raw = db;
#pragma unroll
    for (int r = 0; r < DTRANK; ++r) raw += dw[r] * xr[r];
    const float delta = (raw > 20.0f) ? raw : log1pf(__expf(raw));  // softplus

    const float uu = *up;
    const float zz = *zp;
    const float du = delta * uu;

    float y = 0.0f;
#pragma unroll
    for (int nn = 0; nn < DSTATE; ++nn) {
      const float dA = __expf(delta * An[nn]);
      s[nn] = dA * s[nn] + du * xr[DTRANK + nn];       // B_m at cols [8,24)
      y += s[nn] * xr[DTRANK + DSTATE + nn];           // C_m at cols [24,40)
    }
    y += uu * Dk;
    y *= zz / (1.0f + __expf(-zz));                    // gate: silu(z)
    *yp = y;

    xr += XDBL_LD;
    up += DINNER;
    zp += 2 * DINNER;
    yp += DINNER;
  }
}

// ---------------------------------------------------------------------------
// Mean-pool over L, then latent projection: out = pooled @ W_lat + b_lat.
// Single block of 512 threads (4 batches x 128 dims), pooled staged in LDS.
// ---------------------------------------------------------------------------
__global__ void pool_lat_kernel(const float* __restrict__ h,    // [M, DMODEL]
                                const float* __restrict__ Wl,   // [DMODEL, 128]
                                const float* __restrict__ bl,   // [128]
                                float* __restrict__ out) {      // [B, 128]
  __shared__ float pooled[B_SZ][DMODEL];
  const int t  = threadIdx.x;                // 0..511
  const int b  = t >> 7;
  const int dm = t & 127;

  float acc = 0.0f;
  const float* hp = h + (size_t)b * L_SZ * DMODEL + dm;
  for (int l = 0; l < L_SZ; ++l) acc += hp[(size_t)l * DMODEL];
  pooled[b][dm] = acc * (1.0f / (float)L_SZ);
  __syncthreads();

  float o = bl[dm];
#pragma unroll 4
  for (int dd = 0; dd < DMODEL; ++dd)
    o += pooled[b][dd] * Wl[dd * 128 + dm];
  out[b * 128 + dm] = o;
}

// ---------------------------------------------------------------------------
extern "C" void kernel_launch(void* const* d_in, const int* in_sizes, int n_in,
                              void* d_out, int out_size, void* d_ws, size_t ws_size,
                              hipStream_t stream) {
  (void)in_sizes; (void)n_in; (void)out_size; (void)ws_size;

  const float* x          = (const float*)d_in[0];   // [4,2048,64]
  const float* W_emb      = (const float*)d_in[1];   // [64,128]
  const float* b_emb      = (const float*)d_in[2];   // [128]
  const float* in_proj_w  = (const float*)d_in[3];   // [4,512,128]
  const float* conv_w     = (const float*)d_in[4];   // [4,256,4]
  const float* conv_b     = (const float*)d_in[5];   // [4,256]
  const float* x_proj_w   = (const float*)d_in[6];   // [4,40,256]
  const float* dt_proj_w  = (const float*)d_in[7];   // [4,256,8]
  const float* dt_proj_b  = (const float*)d_in[8];   // [4,256]
  const float* A_log      = (const float*)d_in[9];   // [4,256,16]
  const float* D_skip     = (const float*)d_in[10];  // [4,256]
  const float* out_proj_w = (const float*)d_in[11];  // [4,128,256]
  const float* W_lat      = (const float*)d_in[12];  // [128,128]
  const float* b_lat      = (const float*)d_in[13];  // [128]
  float* out = (float*)d_out;

  // Workspace layout (floats): h | xz | u | xdbl | y   (~39.3 MB total)
  float* ws   = (float*)d_ws;
  float* h    = ws;
  float* xz   = h   + (size_t)M_ROWS * DMODEL;        // 1,048,576
  float* u    = xz  + (size_t)M_ROWS * 2 * DINNER;    // 4,194,304
  float* xd   = u   + (size_t)M_ROWS * DINNER;        // 2,097,152
  float* y    = xd  + (size_t)M_ROWS * XDBL_LD;       //   393,216

  const dim3 blk(256);
  const int WPB = 256 / 32;  // waves per block

  // ---- embedding: h = x @ W_emb + b_emb   (B stored [K, N], NB=2)
  {
    const int wgroups = (M_ROWS / 16) * ((DMODEL / 16) / 2);
    gemm_f32_wmma<0, DMODEL, DMODEL, IN_DIM, 2, true>
        <<<(wgroups + WPB - 1) / WPB, blk, 0, stream>>>(x, W_emb, b_emb, h);
  }

  for (int i = 0; i < 4; ++i) {
    // ---- in_proj: xz = h @ in_w^T  [8192 x 512], NB=2
    {
      const int wgroups = (M_ROWS / 16) * (((2 * DINNER) / 16) / 2);
      gemm_f32_wmma<1, 2 * DINNER, 2 * DINNER, DMODEL, 2, false>
          <<<(wgroups + WPB - 1) / WPB, blk, 0, stream>>>(
              h, in_proj_w + (size_t)i * 2 * DINNER * DMODEL, nullptr, xz);
    }
    // ---- depthwise causal conv + silu -> u
    {
      const int total = M_ROWS * DINNER;
      conv_silu_kernel<<<(total + 255) / 256, blk, 0, stream>>>(
          xz, conv_w + (size_t)i * DINNER * 4, conv_b + (size_t)i * DINNER, u);
    }
    // ---- x_proj: xdbl = u @ xp_w^T  [8192 x 40] (pad 48), NB=1
    {
      const int wgroups = (M_ROWS / 16) * (XDBL_LD / 16);
      gemm_f32_wmma<1, DTRANK + 2 * DSTATE, XDBL_LD, DINNER, 1, false>
          <<<(wgroups + WPB - 1) / WPB, blk, 0, stream>>>(
              u, x_proj_w + (size_t)i * (DTRANK + 2 * DSTATE) * DINNER, nullptr, xd);
    }
    // ---- fused selective scan -> y
    {
      scan_kernel<<<(B_SZ * DINNER) / 256, blk, 0, stream>>>(
          xd, u, xz,
          dt_proj_w + (size_t)i * DINNER * DTRANK,
          dt_proj_b + (size_t)i * DINNER,
          A_log     + (size_t)i * DINNER * DSTATE,
          D_skip    + (size_t)i * DINNER, y);
    }
    // ---- out_proj: h = y @ out_w^T  [8192 x 128], NB=2
    {
      const int wgroups = (M_ROWS / 16) * ((DMODEL / 16) / 2);
      gemm_f32_wmma<1, DMODEL, DMODEL, DINNER, 2, false>
          <<<(wgroups + WPB - 1) / WPB, blk, 0, stream>>>(
              y, out_proj_w + (size_t)i * DMODEL * DINNER, nullptr, h);
    }
  }

  // ---- mean pool + latent projection
  pool_lat_kernel<<<1, 512, 0, stream>>>(h, W_lat, b_lat, out);
}